// GATLayer_20186346291918
// MI455X (gfx1250) — compile-verified
//
#include <hip/hip_runtime.h>
#include <math.h>

typedef __attribute__((ext_vector_type(2))) float v2f;
typedef __attribute__((ext_vector_type(4))) float v4f;
typedef __attribute__((ext_vector_type(8))) float v8f;

#define BB 8
#define NN 1024
#define DD 256
#define HH 4
#define HD 64

// -------- Kernel 1: proj = x @ W^T  (M=8192, N=256, K=256), fp32 WMMA 16x16x4.
// One 16x16 C tile per wave; 4 waves/block cover a 16x64 strip (A reuse via L0/L2).
// A fragment (16x4 f32): lane<16 holds M=lane,K={k0,k0+1}; lane>=16 holds K={k0+2,k0+3}.
__global__ __launch_bounds__(128) void proj_gemm(const float* __restrict__ X,
                                                 const float* __restrict__ W,
                                                 float* __restrict__ PROJ) {
  const int wave = threadIdx.x >> 5;
  const int lane = threadIdx.x & 31;
  const int half = lane >> 4;
  const int lm   = lane & 15;
  const int row0 = blockIdx.x * 16;
  const int col0 = blockIdx.y * 64 + wave * 16;

  v8f c = {};
  const float* arow = X + (size_t)(row0 + lm) * DD + 2 * half;  // x[row][k]
  const float* brow = W + (size_t)(col0 + lm) * DD + 2 * half;  // W[col][k] (C=X*W^T)
#pragma unroll 4
  for (int k0 = 0; k0 < DD; k0 += 4) {
    v2f a = *(const v2f*)(arow + k0);
    v2f b = *(const v2f*)(brow + k0);
    c = __builtin_amdgcn_wmma_f32_16x16x4_f32(false, a, false, b, (short)0, c, false, false);
  }
  float* crow = PROJ + col0 + lm;
#pragma unroll
  for (int r = 0; r < 8; ++r)
    crow[(size_t)(row0 + r + 8 * half) * DD] = c[r];
}

// -------- Kernel 2: s_l[bn,h] = proj[bn,h,:] . w_l ; s_r likewise (VALU, tiny).
__global__ __launch_bounds__(256) void head_scores(const float* __restrict__ PROJ,
                                                   const float* __restrict__ SW,
                                                   float* __restrict__ SL,
                                                   float* __restrict__ SR) {
  int t = blockIdx.x * blockDim.x + threadIdx.x;  // over B*N*H
  int h  = t & (HH - 1);
  int bn = t >> 2;
  const float* p = PROJ + (size_t)bn * DD + h * HD;
  float sl = 0.f, sr = 0.f;
#pragma unroll
  for (int i = 0; i < HD; i += 4) {
    v4f pv = *(const v4f*)(p + i);
    v4f wl = *(const v4f*)(SW + i);
    v4f wr = *(const v4f*)(SW + HD + i);
    sl += pv.x * wl.x + pv.y * wl.y + pv.z * wl.z + pv.w * wl.w;
    sr += pv.x * wr.x + pv.y * wr.y + pv.z * wr.z + pv.w * wr.w;
  }
  SL[t] = sl;
  SR[t] = sr;
}

// -------- Kernel 3: fused masked-softmax attention + msg GEMM + epilogue.
// Block = 4 waves = 4 heads for one (b, 16-query-row) tile. Flash-style loop over m.
// Score tile lives in WMMA C layout: VGPR r <-> row r+8*half, lanes lm <-> col.
__global__ __launch_bounds__(128) void gat_attn(const float* __restrict__ X,
                                                const float* __restrict__ ADJ,
                                                const float* __restrict__ PROJ,
                                                const float* __restrict__ SL,
                                                const float* __restrict__ SR,
                                                const float* __restrict__ SW,
                                                const float* __restrict__ BIAS,
                                                float* __restrict__ OUT) {
  __shared__ float ldsP[HH * 16 * 17];  // per-wave 16x16 transpose pad, stride 17 (conflict-free)
  const int h    = threadIdx.x >> 5;    // wave id == head
  const int lane = threadIdx.x & 31;
  const int half = lane >> 4;
  const int lm   = lane & 15;
  const int b    = blockIdx.x >> 6;
  const int n0   = (blockIdx.x & 63) * 16;
  const float wa = SW[2 * HD];
  float* myP = ldsP + h * (16 * 17);

  float slA[8];
#pragma unroll
  for (int r = 0; r < 8; ++r)
    slA[r] = SL[(size_t)(b * NN + n0 + r + 8 * half) * HH + h];

  float mrow[8], lrow[8];
  v8f acc[4];
  v8f vzero = {};
#pragma unroll
  for (int r = 0; r < 8; ++r) { mrow[r] = -1e30f; lrow[r] = 0.f; }
#pragma unroll
  for (int c = 0; c < 4; ++c) acc[c] = vzero;

  const float* adjBase = ADJ + (size_t)(b * NN + n0) * NN;

  for (int m0 = 0; m0 < NN; m0 += 16) {
    float srj = SR[(size_t)(b * NN + m0 + lm) * HH + h];
    float pfrag[8];
#pragma unroll
    for (int r = 0; r < 8; ++r) {
      float a = adjBase[(size_t)(r + 8 * half) * NN + m0 + lm];
      float s = slA[r] + srj + a * wa;
      s = s > 0.f ? s : 0.2f * s;            // leaky_relu(0.2)
      if (a <= 1e-5f) s -= 9e15f;            // mask
      // row max across the 16-lane group holding this row
      float mx = s;
      mx = fmaxf(mx, __shfl_xor(mx, 1, 32));
      mx = fmaxf(mx, __shfl_xor(mx, 2, 32));
      mx = fmaxf(mx, __shfl_xor(mx, 4, 32));
      mx = fmaxf(mx, __shfl_xor(mx, 8, 32));
      float newm  = fmaxf(mrow[r], mx);
      float alpha = __expf(mrow[r] - newm);
      float p     = __expf(s - newm);
      float ps = p;
      ps += __shfl_xor(ps, 1, 32);
      ps += __shfl_xor(ps, 2, 32);
      ps += __shfl_xor(ps, 4, 32);
      ps += __shfl_xor(ps, 8, 32);
      lrow[r] = lrow[r] * alpha + ps;
      mrow[r] = newm;
#pragma unroll
      for (int c = 0; c < 4; ++c) acc[c][r] *= alpha;
      pfrag[r] = p;
    }
    // transpose P tile into LDS: store column-major (j=lm) so reads give A-layout
#pragma unroll
    for (int r = 0; r < 8; ++r)
      myP[lm * 17 + r + 8 * half] = pfrag[r];
    asm volatile("s_wait_dscnt 0x0" ::: "memory");  // same-wave LDS store->load

    const float* pb = PROJ + (size_t)(b * NN + m0) * DD + h * HD;
#pragma unroll
    for (int kk = 0; kk < 4; ++kk) {
      const int k0 = 4 * kk + 2 * half;
      v2f a;
      a.x = myP[k0 * 17 + lm];
      a.y = myP[(k0 + 1) * 17 + lm];
#pragma unroll
      for (int c = 0; c < 4; ++c) {
        v2f bf;
        bf.x = pb[(size_t)k0 * DD + c * 16 + lm];
        bf.y = pb[(size_t)(k0 + 1) * DD + c * 16 + lm];
        acc[c] = __builtin_amdgcn_wmma_f32_16x16x4_f32(false, a, false, bf, (short)0,
                                                       acc[c], false, false);
      }
    }
  }

  // epilogue: out = leaky_relu(x + msg/l + bias, 0.5)
#pragma unroll
  for (int r = 0; r < 8; ++r) {
    float inv = 1.f / lrow[r];
    int row = n0 + r + 8 * half;
    size_t base = (size_t)(b * NN + row) * DD + h * HD;
#pragma unroll
    for (int c = 0; c < 4; ++c) {
      int d = h * HD + c * 16 + lm;
      float o = X[base + c * 16 + lm] + acc[c][r] * inv + BIAS[d];
      o = o > 0.f ? o : 0.5f * o;
      OUT[base + c * 16 + lm] = o;
    }
  }
}

extern "C" void kernel_launch(void* const* d_in, const int* in_sizes, int n_in,
                              void* d_out, int out_size, void* d_ws, size_t ws_size,
                              hipStream_t stream) {
  const float* x    = (const float*)d_in[0];
  const float* adj  = (const float*)d_in[1];
  const float* W    = (const float*)d_in[2];
  const float* sw   = (const float*)d_in[3];
  const float* bias = (const float*)d_in[4];
  float* out  = (float*)d_out;
  float* proj = (float*)d_ws;                       // 8 MB
  float* sl   = proj + (size_t)BB * NN * DD;        // 128 KB
  float* sr   = sl + (size_t)BB * NN * HH;          // 128 KB

  // second tuple output: adj passthrough
  hipMemcpyAsync(out + (size_t)BB * NN * DD, adj,
                 (size_t)BB * NN * NN * sizeof(float),
                 hipMemcpyDeviceToDevice, stream);

  proj_gemm<<<dim3((BB * NN) / 16, DD / 64), 128, 0, stream>>>(x, W, proj);
  head_scores<<<(BB * NN * HH) / 256, 256, 0, stream>>>(proj, sw, sl, sr);
  gat_attn<<<BB * (NN / 16), 128, 0, stream>>>(x, adj, proj, sl, sr, sw, bias, out);
}